// GraphEncoder_42099269436239
// MI455X (gfx1250) — compile-verified
//
#include <hip/hip_runtime.h>

#define N_NODES 50000
#define N_EDGES 800000
#define ESC 50
#define NLAYERS 2

typedef __attribute__((ext_vector_type(2))) float v2f;
typedef __attribute__((ext_vector_type(8))) float v8f;

#define INV_SQRT_DIN 0.091669846f     // 1/sqrt(119)
#define TP_NORM      0.125f           // 1/sqrt(64)
#define INV_SQRT32   0.17677669529663687f
#define SQRT3        1.7320508075688772f
#define INV_SQRT3    0.5773502691896258f

__device__ __forceinline__ float sigm(float x) { return 1.0f / (1.0f + __expf(-x)); }

__device__ __forceinline__ float warp_sum(float v) {
#pragma unroll
  for (int m = 16; m >= 1; m >>= 1) v += __shfl_xor(v, m, 32);
  return v;
}

__device__ __forceinline__ v8f wmma4(v2f a, v2f b, v8f c) {
  return __builtin_amdgcn_wmma_f32_16x16x4_f32(false, a, false, b, (short)0, c, false, false);
}

// C[16x16] += A(LDS,row-major,lda)[16xK=32] * B(global,K-major,ldn)[32x16 at col n0]
__device__ __forceinline__ v8f gemm_k32(const float* A, int lda, const float* B, int ldn,
                                        int n0, v8f acc) {
  const int lane = threadIdx.x & 31;
  const int r = lane & 15;
  const int kh = (lane >> 4) << 1;          // 0 or 2
  const int col = n0 + r;
#pragma unroll
  for (int k0 = 0; k0 < 32; k0 += 4) {
    v2f a, b;
    a.x = A[r * lda + k0 + kh];
    a.y = A[r * lda + k0 + kh + 1];
    b.x = B[(k0 + kh) * ldn + col];
    b.y = B[(k0 + kh + 1) * ldn + col];
    acc = wmma4(a, b, acc);
  }
  return acc;
}

// K=50 version: A is zero-padded to 52 cols in LDS, so for k>=50 the product is
// zero no matter what B holds -> just clamp the B row index (no masking/branches).
__device__ __forceinline__ v8f gemm_k50(const float* A, int lda, const float* B, int ldn,
                                        int n0, v8f acc) {
  const int lane = threadIdx.x & 31;
  const int r = lane & 15;
  const int kh = (lane >> 4) << 1;
  const int col = n0 + r;
#pragma unroll
  for (int k0 = 0; k0 < 52; k0 += 4) {
    v2f a, b;
    int k1 = k0 + kh, k2 = k0 + kh + 1;
    int kb1 = (k1 < ESC) ? k1 : (ESC - 1);  // clamped: A pad is zero
    int kb2 = (k2 < ESC) ? k2 : (ESC - 1);
    a.x = A[r * lda + k1];
    a.y = A[r * lda + k2];
    b.x = B[kb1 * ldn + col];
    b.y = B[kb2 * ldn + col];
    acc = wmma4(a, b, acc);
  }
  return acc;
}

__global__ __launch_bounds__(256) void k_zero(float* p, int n) {
  int i = blockIdx.x * 256 + threadIdx.x;
  if (i < n) p[i] = 0.0f;
}

// ---------------- embed: one wave per node ----------------
__global__ __launch_bounds__(256) void k_embed(
    const float* __restrict__ x, const float* __restrict__ deg, const float* __restrict__ pos,
    const float* __restrict__ Ws, const float* __restrict__ Wv,
    const float* __restrict__ lng, const float* __restrict__ lnb,
    const float* __restrict__ Gw, const float* __restrict__ Gb,
    float* __restrict__ xs, float* __restrict__ xv) {
  int wave = threadIdx.x >> 5, lane = threadIdx.x & 31;
  int n = blockIdx.x * 8 + wave;
  if (n >= N_NODES) return;
  const float* xr = x + (long)n * 118;
  float acc = 0.0f;
  for (int k = 0; k < 118; ++k) acc = fmaf(xr[k], Ws[k * 32 + lane], acc);
  acc = fmaf(deg[n], Ws[118 * 32 + lane], acc);
  acc *= INV_SQRT_DIN;
  // LayerNorm + SiLU (stage 0)
  float m = warp_sum(acc) * (1.0f / 32.0f);
  float d = acc - m;
  float var = warp_sum(d * d) * (1.0f / 32.0f);
  float z = d * rsqrtf(var + 1e-5f) * lng[lane] + lnb[lane];
  float s = z * sigm(z);
  // gate
  float g = Gb[lane];
  for (int c = 0; c < 32; ++c) g = fmaf(__shfl(s, c, 32), Gw[c * 32 + lane], g);
  g = sigm(g);
  xs[(long)n * 32 + lane] = s;
  float wv = Wv[lane] * g;
  float px = pos[n * 3 + 0], py = pos[n * 3 + 1], pz = pos[n * 3 + 2];
  long o = (long)n * 96 + lane * 3;
  xv[o + 0] = px * wv;
  xv[o + 1] = py * wv;
  xv[o + 2] = pz * wv;
}

// ---------------- edge kernel: one wave per 16-edge tile ----------------
#define EW_PER 5648
__global__ __launch_bounds__(64) void k_edge(
    const float* __restrict__ xs, const float* __restrict__ xv,
    const int* __restrict__ eidx, const float* __restrict__ evec, const float* __restrict__ elen,
    const float* __restrict__ Wss, const float* __restrict__ Wvs,
    const float* __restrict__ Wsv, const float* __restrict__ Wvv,
    const float* __restrict__ Rw, const float* __restrict__ Rb,
    float* __restrict__ agg_s, float* __restrict__ agg_v) {
  __shared__ float sm[2 * EW_PER];
  const int wave = threadIdx.x >> 5, lane = threadIdx.x & 31;
  float* base = sm + wave * EW_PER;
  float* ms  = base;          // 16x33
  float* mv0 = base + 528;    // 16x33 per cartesian component
  float* mv1 = base + 1056;
  float* mv2 = base + 1584;
  float* dt  = base + 2112;   // 16x33
  float* eln = base + 2640;   // 16x53 (cols 50,51 zero)
  float* scl = base + 3488;   // 16x129
  float* yv  = base + 5552;   // 16x4 : Y1 = sqrt3*rhat
  int* srcs = (int*)(base + 5616);
  int* dsts = srcs + 16;
  const int tile = blockIdx.x * 2 + wave;
  const int e0 = tile * 16;

  if (lane < 16) {
    int e = e0 + lane;
    srcs[lane] = eidx[e];
    dsts[lane] = eidx[N_EDGES + e];
    float vx = evec[e * 3], vy = evec[e * 3 + 1], vz = evec[e * 3 + 2];
    float rn = rsqrtf(vx * vx + vy * vy + vz * vz) * SQRT3;
    yv[lane * 4 + 0] = vx * rn;
    yv[lane * 4 + 1] = vy * rn;
    yv[lane * 4 + 2] = vz * rn;
  }
  // gather scalars / vectors (per-wave LDS, lockstep => no barrier needed)
  for (int i = lane; i < 512; i += 32) {
    int r = i >> 5, c = i & 31;
    ms[r * 33 + c] = xs[(long)srcs[r] * 32 + c];
  }
  for (int i = lane; i < 1536; i += 32) {
    int r = i / 96, j = i % 96;
    float val = xv[(long)srcs[r] * 96 + j];
    int v = j / 3, c = j % 3;
    (c == 0 ? mv0 : c == 1 ? mv1 : mv2)[r * 33 + v] = val;
  }
  for (int i = lane; i < 512; i += 32) {
    int r = i >> 5, v = i & 31;
    dt[r * 33 + v] = (mv0[r * 33 + v] * yv[r * 4 + 0] + mv1[r * 33 + v] * yv[r * 4 + 1] +
                      mv2[r * 33 + v] * yv[r * 4 + 2]) * INV_SQRT3;
  }
  for (int i = lane; i < 16 * 53; i += 32) {
    int r = i / 53, c = i % 53;
    eln[i] = (c < ESC) ? elen[(long)(e0 + r) * ESC + c] : 0.0f;
  }
  const int r16 = lane & 15;
  // radial gating: sigmoid([16,50] @ [50,128] + b)
#pragma unroll
  for (int nt = 0; nt < 8; ++nt) {
    v8f acc = {0.f, 0.f, 0.f, 0.f, 0.f, 0.f, 0.f, 0.f};
    acc = gemm_k50(eln, 53, Rw, 128, nt * 16, acc);
    float bb = Rb[nt * 16 + r16];
#pragma unroll
    for (int r = 0; r < 8; ++r) {
      int row = (lane < 16) ? r : r + 8;
      scl[row * 129 + nt * 16 + r16] = sigm(acc[r] + bb);
    }
  }
  // out_s = (ms@Wss + dot@Wvs)*TP_NORM*scale -> scatter
#pragma unroll
  for (int nt = 0; nt < 2; ++nt) {
    v8f acc = {0.f, 0.f, 0.f, 0.f, 0.f, 0.f, 0.f, 0.f};
    acc = gemm_k32(ms, 33, Wss, 32, nt * 16, acc);
    acc = gemm_k32(dt, 33, Wvs, 32, nt * 16, acc);
#pragma unroll
    for (int r = 0; r < 8; ++r) {
      int row = (lane < 16) ? r : r + 8;
      int n = nt * 16 + r16;
      float o = acc[r] * TP_NORM * scl[row * 129 + n];
      atomicAdd(&agg_s[(long)dsts[row] * 32 + n], o);
    }
  }
  // sv = ms@Wsv (kept in accumulators)
  v8f sv0 = {0.f, 0.f, 0.f, 0.f, 0.f, 0.f, 0.f, 0.f};
  v8f sv1 = {0.f, 0.f, 0.f, 0.f, 0.f, 0.f, 0.f, 0.f};
  sv0 = gemm_k32(ms, 33, Wsv, 32, 0, sv0);
  sv1 = gemm_k32(ms, 33, Wsv, 32, 16, sv1);
  // out_v = (sv*Y1 + mv@Wvv)*TP_NORM*scale -> scatter
#pragma unroll
  for (int c = 0; c < 3; ++c) {
    const float* mvc = (c == 0) ? mv0 : (c == 1) ? mv1 : mv2;
#pragma unroll
    for (int nt = 0; nt < 2; ++nt) {
      v8f acc = {0.f, 0.f, 0.f, 0.f, 0.f, 0.f, 0.f, 0.f};
      acc = gemm_k32(mvc, 33, Wvv, 32, nt * 16, acc);
      v8f sv = nt ? sv1 : sv0;
#pragma unroll
      for (int r = 0; r < 8; ++r) {
        int row = (lane < 16) ? r : r + 8;
        int n = nt * 16 + r16;
        float o = (sv[r] * yv[row * 4 + c] + acc[r]) * TP_NORM * scl[row * 129 + 32 + n * 3 + c];
        atomicAdd(&agg_v[(long)dsts[row] * 96 + n * 3 + c], o);
      }
    }
  }
}

// ---------------- node update: one wave per 16-node tile ----------------
__global__ __launch_bounds__(128) void k_node(
    float* __restrict__ xs, float* __restrict__ xv,
    const float* __restrict__ agg_s, const float* __restrict__ agg_v,
    const float* __restrict__ Ls, const float* __restrict__ Lv,
    const float* __restrict__ lng, const float* __restrict__ lnb,
    const float* __restrict__ Gw, const float* __restrict__ Gb) {
  __shared__ float sm[4 * 2640];
  int wave = threadIdx.x >> 5, lane = threadIdx.x & 31;
  int tile = blockIdx.x * 4 + wave;
  if (tile >= N_NODES / 16) return;
  float* base = sm + wave * 2640;
  float* hs = base;
  float* hv0 = base + 528;
  float* hv1 = base + 1056;
  float* hv2 = base + 1584;
  float* sb = base + 2112;
  int g0 = tile * 16;
  for (int i = lane; i < 512; i += 32) {
    int r = i >> 5, c = i & 31;
    long o = (long)(g0 + r) * 32 + c;
    hs[r * 33 + c] = xs[o] + agg_s[o];
  }
  for (int i = lane; i < 1536; i += 32) {
    int r = i / 96, j = i % 96;
    long o = (long)(g0 + r) * 96 + j;
    float val = xv[o] + agg_v[o];
    int v = j / 3, c = j % 3;
    (c == 0 ? hv0 : c == 1 ? hv1 : hv2)[r * 33 + v] = val;
  }
  const int r16 = lane & 15;
#pragma unroll
  for (int nt = 0; nt < 2; ++nt) {
    v8f acc = {0.f, 0.f, 0.f, 0.f, 0.f, 0.f, 0.f, 0.f};
    acc = gemm_k32(hs, 33, Ls, 32, nt * 16, acc);
#pragma unroll
    for (int r = 0; r < 8; ++r) {
      int row = (lane < 16) ? r : r + 8;
      sb[row * 33 + nt * 16 + r16] = acc[r] * INV_SQRT32;
    }
  }
#pragma unroll
  for (int c = 0; c < 3; ++c) {
    float* hv = (c == 0) ? hv0 : (c == 1) ? hv1 : hv2;
    v8f a0 = {0.f, 0.f, 0.f, 0.f, 0.f, 0.f, 0.f, 0.f};
    v8f a1 = {0.f, 0.f, 0.f, 0.f, 0.f, 0.f, 0.f, 0.f};
    a0 = gemm_k32(hv, 33, Lv, 32, 0, a0);
    a1 = gemm_k32(hv, 33, Lv, 32, 16, a1);
#pragma unroll
    for (int r = 0; r < 8; ++r) {
      int row = (lane < 16) ? r : r + 8;
      hv[row * 33 + r16] = a0[r] * INV_SQRT32;
      hv[row * 33 + 16 + r16] = a1[r] * INV_SQRT32;
    }
  }
  float lg = lng[lane], lb = lnb[lane];
  for (int r = 0; r < 16; ++r) {
    float s = sb[r * 33 + lane];
    float m = warp_sum(s) * (1.0f / 32.0f);
    float d = s - m;
    float var = warp_sum(d * d) * (1.0f / 32.0f);
    float z = d * rsqrtf(var + 1e-5f) * lg + lb;
    sb[r * 33 + lane] = z * sigm(z);
  }
  for (int i = lane; i < 512; i += 32) {
    int r = i >> 5, c = i & 31;
    xs[(long)(g0 + r) * 32 + c] = sb[r * 33 + c];
  }
#pragma unroll
  for (int nt = 0; nt < 2; ++nt) {
    v8f acc = {0.f, 0.f, 0.f, 0.f, 0.f, 0.f, 0.f, 0.f};
    acc = gemm_k32(sb, 33, Gw, 32, nt * 16, acc);
    float gb = Gb[nt * 16 + r16];
#pragma unroll
    for (int r = 0; r < 8; ++r) {
      int row = (lane < 16) ? r : r + 8;
      int v = nt * 16 + r16;
      float g = sigm(acc[r] + gb);
      long o = (long)(g0 + row) * 96 + v * 3;
      xv[o + 0] = hv0[row * 33 + v] * g;
      xv[o + 1] = hv1[row * 33 + v] * g;
      xv[o + 2] = hv2[row * 33 + v] * g;
    }
  }
}

// ---------------- final: output linears + tokens ----------------
__global__ __launch_bounds__(64) void k_final(
    const float* __restrict__ xs, const float* __restrict__ xv,
    const float* __restrict__ Os, const float* __restrict__ Ov,
    const float* __restrict__ lng, const float* __restrict__ lnb,
    const float* __restrict__ Gw, const float* __restrict__ Gb,
    const float* __restrict__ Psw, const float* __restrict__ Psb,
    const float* __restrict__ Pvw, const float* __restrict__ Pvb,
    float* __restrict__ out) {
  __shared__ float sm[2 * 3680];
  int wave = threadIdx.x >> 5, lane = threadIdx.x & 31;
  int tile = blockIdx.x * 2 + wave;
  if (tile >= N_NODES / 16) return;
  float* base = sm + wave * 3680;
  float* hs = base;
  float* hv0 = base + 528;
  float* hv1 = base + 1056;
  float* hv2 = base + 1584;
  float* sb = base + 2112;
  float* sc = base + 2640;  // 16x65
  int g0 = tile * 16;
  for (int i = lane; i < 512; i += 32) {
    int r = i >> 5, c = i & 31;
    hs[r * 33 + c] = xs[(long)(g0 + r) * 32 + c];
  }
  for (int i = lane; i < 1536; i += 32) {
    int r = i / 96, j = i % 96;
    float val = xv[(long)(g0 + r) * 96 + j];
    int v = j / 3, c = j % 3;
    (c == 0 ? hv0 : c == 1 ? hv1 : hv2)[r * 33 + v] = val;
  }
  const int r16 = lane & 15;
#pragma unroll
  for (int nt = 0; nt < 2; ++nt) {
    v8f acc = {0.f, 0.f, 0.f, 0.f, 0.f, 0.f, 0.f, 0.f};
    acc = gemm_k32(hs, 33, Os, 32, nt * 16, acc);
#pragma unroll
    for (int r = 0; r < 8; ++r) {
      int row = (lane < 16) ? r : r + 8;
      sb[row * 33 + nt * 16 + r16] = acc[r] * INV_SQRT32;
    }
  }
#pragma unroll
  for (int c = 0; c < 3; ++c) {
    float* hv = (c == 0) ? hv0 : (c == 1) ? hv1 : hv2;
    v8f a0 = {0.f, 0.f, 0.f, 0.f, 0.f, 0.f, 0.f, 0.f};
    v8f a1 = {0.f, 0.f, 0.f, 0.f, 0.f, 0.f, 0.f, 0.f};
    a0 = gemm_k32(hv, 33, Ov, 32, 0, a0);
    a1 = gemm_k32(hv, 33, Ov, 32, 16, a1);
#pragma unroll
    for (int r = 0; r < 8; ++r) {
      int row = (lane < 16) ? r : r + 8;
      hv[row * 33 + r16] = a0[r] * INV_SQRT32;
      hv[row * 33 + 16 + r16] = a1[r] * INV_SQRT32;
    }
  }
  float lg = lng[lane], lb = lnb[lane];
  for (int r = 0; r < 16; ++r) {
    float s = sb[r * 33 + lane];
    float m = warp_sum(s) * (1.0f / 32.0f);
    float d = s - m;
    float var = warp_sum(d * d) * (1.0f / 32.0f);
    float z = d * rsqrtf(var + 1e-5f) * lg + lb;
    sb[r * 33 + lane] = z * sigm(z);
  }
#pragma unroll
  for (int nt = 0; nt < 2; ++nt) {
    v8f acc = {0.f, 0.f, 0.f, 0.f, 0.f, 0.f, 0.f, 0.f};
    acc = gemm_k32(sb, 33, Gw, 32, nt * 16, acc);
    float gb = Gb[nt * 16 + r16];
#pragma unroll
    for (int r = 0; r < 8; ++r) {
      int row = (lane < 16) ? r : r + 8;
      int v = nt * 16 + r16;
      float g = sigm(acc[r] + gb);
      hv0[row * 33 + v] *= g;
      hv1[row * 33 + v] *= g;
      hv2[row * 33 + v] *= g;
    }
  }
  // sc = s @ Ps_w + Ps_b  (store to LDS)
#pragma unroll
  for (int ht = 0; ht < 4; ++ht) {
    v8f acc = {0.f, 0.f, 0.f, 0.f, 0.f, 0.f, 0.f, 0.f};
    acc = gemm_k32(sb, 33, Psw, 64, ht * 16, acc);
    float bb = Psb[ht * 16 + r16];
#pragma unroll
    for (int r = 0; r < 8; ++r) {
      int row = (lane < 16) ? r : r + 8;
      sc[row * 65 + ht * 16 + r16] = acc[r] + bb;
    }
  }
  // tok[n,axis,h] = xv_axis @ Pv_w + Pv_b + sc
#pragma unroll
  for (int c = 0; c < 3; ++c) {
    const float* hv = (c == 0) ? hv0 : (c == 1) ? hv1 : hv2;
#pragma unroll
    for (int ht = 0; ht < 4; ++ht) {
      v8f acc = {0.f, 0.f, 0.f, 0.f, 0.f, 0.f, 0.f, 0.f};
      acc = gemm_k32(hv, 33, Pvw, 64, ht * 16, acc);
      float bb = Pvb[ht * 16 + r16];
#pragma unroll
      for (int r = 0; r < 8; ++r) {
        int row = (lane < 16) ? r : r + 8;
        int h = ht * 16 + r16;
        out[((long)(g0 + row) * 3 + c) * 64 + h] = acc[r] + bb + sc[row * 65 + h];
      }
    }
  }
}

extern "C" void kernel_launch(void* const* d_in, const int* in_sizes, int n_in,
                              void* d_out, int out_size, void* d_ws, size_t ws_size,
                              hipStream_t stream) {
  const float* x    = (const float*)d_in[0];
  const float* deg  = (const float*)d_in[1];
  const float* pos  = (const float*)d_in[2];
  const float* evec = (const float*)d_in[3];
  const float* elen = (const float*)d_in[4];
  const int*   eidx = (const int*)d_in[5];
  const float* Wes  = (const float*)d_in[6];
  const float* Wev  = (const float*)d_in[7];
  const float* Wss  = (const float*)d_in[8];
  const float* Wvs  = (const float*)d_in[9];
  const float* Wsv  = (const float*)d_in[10];
  const float* Wvv  = (const float*)d_in[11];
  const float* Ls   = (const float*)d_in[12];
  const float* Lv   = (const float*)d_in[13];
  const float* Rw   = (const float*)d_in[14];
  const float* Rb   = (const float*)d_in[15];
  const float* lng  = (const float*)d_in[16];
  const float* lnb  = (const float*)d_in[17];
  const float* Gw   = (const float*)d_in[18];
  const float* Gb   = (const float*)d_in[19];
  const float* Os   = (const float*)d_in[20];
  const float* Ov   = (const float*)d_in[21];
  const float* Psw  = (const float*)d_in[22];
  const float* Psb  = (const float*)d_in[23];
  const float* Pvw  = (const float*)d_in[24];
  const float* Pvb  = (const float*)d_in[25];
  float* out = (float*)d_out;

  float* xs    = (float*)d_ws;                 // N*32
  float* xv    = xs + (size_t)N_NODES * 32;    // N*96
  float* agg_s = xv + (size_t)N_NODES * 96;    // N*32
  float* agg_v = agg_s + (size_t)N_NODES * 32; // N*96 (contiguous with agg_s)

  k_embed<<<N_NODES / 8, 256, 0, stream>>>(x, deg, pos, Wes, Wev, lng, lnb, Gw, Gb, xs, xv);

  for (int l = 0; l < NLAYERS; ++l) {
    k_zero<<<(N_NODES * 128 + 255) / 256, 256, 0, stream>>>(agg_s, N_NODES * 128);
    k_edge<<<N_EDGES / 32, 64, 0, stream>>>(xs, xv, eidx, evec, elen,
                                            Wss + l * 1024, Wvs + l * 1024,
                                            Wsv + l * 1024, Wvv + l * 1024,
                                            Rw + l * ESC * 128, Rb + l * 128,
                                            agg_s, agg_v);
    k_node<<<(N_NODES / 16 + 3) / 4, 128, 0, stream>>>(xs, xv, agg_s, agg_v,
                                                       Ls + l * 1024, Lv + l * 1024,
                                                       lng + (l + 1) * 32, lnb + (l + 1) * 32,
                                                       Gw + (l + 1) * 1024, Gb + (l + 1) * 32);
  }

  k_final<<<(N_NODES / 16 + 1) / 2, 64, 0, stream>>>(xs, xv, Os, Ov,
                                                     lng + 3 * 32, lnb + 3 * 32,
                                                     Gw + 3 * 1024, Gb + 3 * 32,
                                                     Psw, Psb, Pvw, Pvb, out);
}